// KNNSearch_22119081574484
// MI455X (gfx1250) — compile-verified
//
#include <hip/hip_runtime.h>
#include <float.h>

// KNN (k=16) on MI455X:
//   - distances via V_WMMA_F32_16X16X4_F32:  D[m][n] = |p_m|^2 - 2 p_m.q_n
//       A row m = ( px, py, pz, |p|^2 ),  B col n = ( -2qx, -2qy, -2qz, 1 )
//   - points chunks staged global->LDS by the Tensor Data Mover (async,
//     double-buffered, s_wait_tensorcnt + barrier), converted to float4 in LDS
//   - D layout puts each lane's 8 results on ONE query => per-lane register
//     top-16 with 1-compare reject; lane pair (n, n+16) merged through LDS.

typedef __attribute__((ext_vector_type(2))) float v2f;
typedef __attribute__((ext_vector_type(8))) float v8f;
typedef __attribute__((ext_vector_type(4))) unsigned int u32x4;
typedef __attribute__((ext_vector_type(8))) int i32x8;
typedef __attribute__((ext_vector_type(4))) int i32x4;

#define KNN_K 16
#define CHUNK_PTS 2048
#define TILES (CHUNK_PTS / 16)
#define BLOCK_THREADS 256
#define WAVES_PER_BLOCK (BLOCK_THREADS / 32)

// ---- Tensor Data Mover: 1-D dword copy global -> LDS ----------------------
__device__ __forceinline__ void tdm_load_dwords(const void* gsrc,
                                                unsigned int lds_byte_addr,
                                                unsigned int ndwords)
{
    const unsigned long long ga = (unsigned long long)(uintptr_t)gsrc;
    u32x4 g0;
    g0[0] = 1u;                                   // count=1, user mode
    g0[1] = lds_byte_addr;                        // lds_addr       [63:32]
    g0[2] = (unsigned int)(ga & 0xFFFFFFFFu);     // global_addr    [95:64]
    g0[3] = (unsigned int)((ga >> 32) & 0x01FFFFFFu) | (2u << 30); // +type=2

    i32x8 g1;
    g1[0] = (int)(2u << 16);                      // data_size=4B; wg_mask=0
    g1[1] = (int)((ndwords & 0xFFFFu) << 16);     // tensor_dim0[15:0] @bit48
    g1[2] = (int)(((ndwords >> 16) & 0xFFFFu)     // tensor_dim0[31:16]
                  | (1u << 16));                  // tensor_dim1 = 1 @bit80
    g1[3] = (int)((ndwords & 0xFFFFu) << 16);     // tile_dim0 @bit112
    g1[4] = 0;                                    // tile_dim1=0, tile_dim2=0
    g1[5] = (int)ndwords;                         // tensor_dim0_stride[31:0]
    g1[6] = 0;
    g1[7] = 0;

    i32x4 gz = {0, 0, 0, 0};
#if defined(__clang_major__) && (__clang_major__ >= 23)
    i32x8 gz8 = {0, 0, 0, 0, 0, 0, 0, 0};
    __builtin_amdgcn_tensor_load_to_lds(g0, g1, gz, gz, gz8, 0);
#else
    __builtin_amdgcn_tensor_load_to_lds(g0, g1, gz, gz, 0);
#endif
}

struct MergeScratch {
    float md[WAVES_PER_BLOCK][16][2 * KNN_K];
    int   mi[WAVES_PER_BLOCK][16][2 * KNN_K];
};

__global__ __launch_bounds__(BLOCK_THREADS)
void knn_wmma_kernel(const float* __restrict__ points,
                     const float* __restrict__ queries,
                     int N, int M,
                     int* __restrict__ out_idx,
                     float* __restrict__ out_dist)
{
    __shared__ float s_raw[2][CHUNK_PTS * 3];     // 48 KB: TDM double buffer
    __shared__ union {
        float4       fmt[CHUNK_PTS];              // 32 KB: (x,y,z,|p|^2)
        MergeScratch mrg;                         // 32 KB: reused after chunks
    } u;

    const int tid  = threadIdx.x;
    const int wave = tid >> 5;
    const int lane = tid & 31;
    const int col  = lane & 15;                   // query column in tile
    const int hi   = lane >> 4;                   // row half / K half selector

    const int qtile = blockIdx.x * WAVES_PER_BLOCK + wave;
    int qn = qtile * 16 + col;
    const bool qvalid = (qn < M);
    if (!qvalid) qn = M - 1;                      // clamp: keep barriers uniform

    const float qx = queries[(size_t)qn * 3 + 0];
    const float qy = queries[(size_t)qn * 3 + 1];
    const float qz = queries[(size_t)qn * 3 + 2];
    const float qsq = qx * qx + qy * qy + qz * qz;

    // B (4x16): v0 = K0 (lanes 0-15) / K2 (lanes 16-31); v1 = K1 / K3
    v2f b;
    b.x = hi ? (-2.0f * qz) : (-2.0f * qx);
    b.y = hi ? 1.0f         : (-2.0f * qy);

    float bd[KNN_K];
    int   bi[KNN_K];
#pragma unroll
    for (int j = 0; j < KNN_K; ++j) { bd[j] = FLT_MAX; bi[j] = 0; }

    const int nchunks = (N + CHUNK_PTS - 1) / CHUNK_PTS;

    // Kick off TDM for chunk 0 (one wave issues; TDM ignores EXEC).
    if (wave == 0) {
        const unsigned int nd0 = (unsigned int)(min(CHUNK_PTS, N) * 3);
        tdm_load_dwords(points, (unsigned int)(uintptr_t)&s_raw[0][0], nd0);
    }

    for (int c = 0; c < nchunks; ++c) {
        const int cur  = c & 1;
        const int base = c * CHUNK_PTS;

        if (wave == 0) __builtin_amdgcn_s_wait_tensorcnt(0);
        __syncthreads();                          // raw[cur] visible to block

        // Overlap: launch TDM for next chunk while we convert + compute.
        if (wave == 0 && c + 1 < nchunks) {
            const int nbase = base + CHUNK_PTS;
            const unsigned int nd =
                (unsigned int)(min(CHUNK_PTS, N - nbase) * 3);
            tdm_load_dwords(points + (size_t)nbase * 3,
                            (unsigned int)(uintptr_t)&s_raw[cur ^ 1][0], nd);
        }

        // LDS->LDS conversion: (x,y,z) -> (x,y,z,|p|^2), sentinel-pad tail.
        for (int i = tid; i < CHUNK_PTS; i += BLOCK_THREADS) {
            float4 v;
            if (base + i < N) {
                const float px = s_raw[cur][i * 3 + 0];
                const float py = s_raw[cur][i * 3 + 1];
                const float pz = s_raw[cur][i * 3 + 2];
                v = make_float4(px, py, pz, px * px + py * py + pz * pz);
            } else {
                v = make_float4(0.f, 0.f, 0.f, 3.0e38f);
            }
            u.fmt[i] = v;
        }
        __syncthreads();                          // fmt ready

        // A fragment t lives at fmt[t+col], halves selected by hi.
        const float2* ap =
            (const float2*)((const char*)&u.fmt[0] + col * 16 + hi * 8);

        // Software pipeline: fetch A(t+1) before WMMA(t).
        float2 acur = ap[0];
#pragma unroll 2
        for (int t = 0; t < TILES; ++t) {
            const int tn = (t + 1 < TILES) ? (t + 1) : t; // uniform (SALU)
            const float2 anxt = ap[2 * tn];

            v2f a; a.x = acur.x; a.y = acur.y;
            v8f acc = {};
            acc = __builtin_amdgcn_wmma_f32_16x16x4_f32(
                false, a, false, b, (short)0, acc, false, false);

            // Min-tree early-out: 1 branch per tile in the steady state.
            const float m0 = fminf(fminf(acc[0], acc[1]), fminf(acc[2], acc[3]));
            const float m1 = fminf(fminf(acc[4], acc[5]), fminf(acc[6], acc[7]));
            if (fminf(m0, m1) < bd[KNN_K - 1]) {
                const int pbase = base + t * 16 + hi * 8;
#pragma unroll
                for (int r = 0; r < 8; ++r) {
                    const float cd = acc[r];
                    if (cd < bd[KNN_K - 1]) {
                        bd[KNN_K - 1] = cd;
                        bi[KNN_K - 1] = pbase + r;
#pragma unroll
                        for (int j = KNN_K - 1; j > 0; --j) {
                            if (bd[j] < bd[j - 1]) {
                                float td = bd[j]; bd[j] = bd[j - 1]; bd[j - 1] = td;
                                int   ti = bi[j]; bi[j] = bi[j - 1]; bi[j - 1] = ti;
                            }
                        }
                    }
                }
            }
            acur = anxt;
        }
        __syncthreads();                          // done with fmt before reuse
    }

    // Lane pair (col, col+16) both served query col: merge sorted lists.
#pragma unroll
    for (int j = 0; j < KNN_K; ++j) {
        u.mrg.md[wave][col][hi * KNN_K + j] = bd[j];
        u.mrg.mi[wave][col][hi * KNN_K + j] = bi[j];
    }
    __syncthreads();

    if (hi == 0 && qvalid) {
        const float* dA = &u.mrg.md[wave][col][0];
        const float* dB = &u.mrg.md[wave][col][KNN_K];
        const int*   iA = &u.mrg.mi[wave][col][0];
        const int*   iB = &u.mrg.mi[wave][col][KNN_K];
        int ia = 0, ib = 0;
        for (int t = 0; t < KNN_K; ++t) {
            const float da = dA[ia], db = dB[ib];
            const int   xa = iA[ia], xb = iB[ib];
            const bool takeA = (da < db) || (da == db && xa < xb); // tie: low idx
            out_dist[(size_t)qn * KNN_K + t] = (takeA ? da : db) + qsq;
            out_idx [(size_t)qn * KNN_K + t] = takeA ? xa : xb;
            ia += takeA ? 1 : 0;
            ib += takeA ? 0 : 1;
        }
    }
}

__global__ void knn_row_splits_kernel(int* __restrict__ rs, int m)
{
    const int i = blockIdx.x * blockDim.x + threadIdx.x;
    if (i <= m) rs[i] = i * KNN_K;
}

extern "C" void kernel_launch(void* const* d_in, const int* in_sizes, int n_in,
                              void* d_out, int out_size, void* d_ws, size_t ws_size,
                              hipStream_t stream)
{
    const float* points  = (const float*)d_in[0];
    const float* queries = (const float*)d_in[1];
    const int N = in_sizes[0] / 3;   // 131072
    const int M = in_sizes[1] / 3;   // 65536

    // d_out: idx [M*k] i32 | row_splits [M+1] i32 | dist [M*k] f32 (flat concat)
    int*   out_idx    = (int*)d_out;
    int*   row_splits = out_idx + (size_t)M * KNN_K;
    float* out_dist   = (float*)(row_splits + (M + 1));

    const int qtiles = (M + 15) / 16;
    const int blocks = (qtiles + WAVES_PER_BLOCK - 1) / WAVES_PER_BLOCK;
    knn_wmma_kernel<<<blocks, BLOCK_THREADS, 0, stream>>>(
        points, queries, N, M, out_idx, out_dist);

    knn_row_splits_kernel<<<(M + 1 + 255) / 256, 256, 0, stream>>>(row_splits, M);
}